// CombineJaggedEmbedding_8358006358636
// MI455X (gfx1250) — compile-verified
//
#include <hip/hip_runtime.h>
#include <stdint.h>
#include <stddef.h>

typedef __attribute__((ext_vector_type(2))) float v2f;
typedef __attribute__((ext_vector_type(4))) float v4f;

#define WAVES_PER_BLOCK 8
#define CHUNK 32   // one index per lane of a wave32

// ---- CDNA5 async global->LDS copy (CDNA5 ISA) ----
// Per-lane: each lane copies one dword from gaddr to its LDS byte address. Tracked by ASYNCcnt (+1 per issue).
__device__ __forceinline__ void async_g2l_b32(uint32_t lds_byte_addr, const void* gaddr) {
    asm volatile("global_load_async_to_lds_b32 %0, %1, off"
                 :: "v"(lds_byte_addr), "v"((uint64_t)(uintptr_t)gaddr)
                 : "memory");
}
__device__ __forceinline__ uint32_t lds_off(const void* p) {
    // flat LDS pointer: addr[31:0] is the LDS byte address (ISA §10.2 aperture rules)
    return (uint32_t)(uintptr_t)p;
}

// =============== feature 0: weighted sum pool, D0 = 64 ===============
__device__ __forceinline__ void pool0_body(
    int bid,
    const float* __restrict__ table0, const float* __restrict__ w0,
    const int* __restrict__ idx0, const int* __restrict__ off0,
    float* __restrict__ out0, int B,
    int (*sIdx)[2][CHUNK], float (*sW)[2][CHUNK])
{
    const int lane = threadIdx.x & 31;
    const int wave = threadIdx.x >> 5;
    const int b    = bid * WAVES_PER_BLOCK + wave;
    if (b >= B) return;

    const int j0 = off0[b];
    const int j1 = off0[b + 1];

    const v2f* __restrict__ t2 = (const v2f*)table0;   // row r at t2[r*32]
    v2f acc = (v2f)0.0f;

    if (j0 < j1) {
        // prologue: stage chunk 0 into LDS asynchronously
        {
            int m = j1 - j0; if (m > CHUNK) m = CHUNK;
            if (lane < m) {
                async_g2l_b32(lds_off(&sIdx[wave][0][lane]), &idx0[j0 + lane]);
                async_g2l_b32(lds_off(&sW[wave][0][lane]),   &w0[j0 + lane]);
            }
        }
        int buf = 0;
        for (int jc = j0; jc < j1; jc += CHUNK, buf ^= 1) {
            int m = j1 - jc; if (m > CHUNK) m = CHUNK;
            const int jn = jc + CHUNK;
            if (jn < j1) {
                // stage next chunk, then wait only for the current one (double buffer)
                int mn = j1 - jn; if (mn > CHUNK) mn = CHUNK;
                if (lane < mn) {
                    async_g2l_b32(lds_off(&sIdx[wave][buf ^ 1][lane]), &idx0[jn + lane]);
                    async_g2l_b32(lds_off(&sW[wave][buf ^ 1][lane]),   &w0[jn + lane]);
                }
                asm volatile("s_wait_asynccnt 0x2" ::: "memory");
            } else {
                asm volatile("s_wait_asynccnt 0x0" ::: "memory");
            }

            // bulk prefetch of the rows this wave is about to gather serially
            if (lane < m) {
                const int pr = sIdx[wave][buf][lane];
                __builtin_prefetch(&t2[(size_t)pr * 32], 0, 0);
            }

            int k = 0;
            for (; k + 8 <= m; k += 8) {
                v2f  v[8];
                float a[8];
                #pragma unroll
                for (int u = 0; u < 8; ++u) {
                    const int r = sIdx[wave][buf][k + u];
                    a[u] = sW[wave][buf][k + u];
                    v[u] = t2[(size_t)r * 32 + lane];
                }
                #pragma unroll
                for (int u = 0; u < 8; ++u) {
                    acc.x = fmaf(a[u], v[u].x, acc.x);
                    acc.y = fmaf(a[u], v[u].y, acc.y);
                }
            }
            for (; k < m; ++k) {
                const int   r = sIdx[wave][buf][k];
                const float a = sW[wave][buf][k];
                const v2f   v = t2[(size_t)r * 32 + lane];
                acc.x = fmaf(a, v.x, acc.x);
                acc.y = fmaf(a, v.y, acc.y);
            }
        }
    }
    // empty segment -> acc == 0 == patch value; NT store keeps tables L2-resident
    __builtin_nontemporal_store(acc, (v2f*)out0 + (size_t)b * 32 + lane);
}

// =============== feature 1: mean pool, D1 = 128 ===============
__device__ __forceinline__ void pool1_body(
    int bid,
    const float* __restrict__ table1,
    const int* __restrict__ idx1, const int* __restrict__ off1,
    float* __restrict__ out1, int B,
    int (*sIdx)[2][CHUNK])
{
    const int lane = threadIdx.x & 31;
    const int wave = threadIdx.x >> 5;
    const int b    = bid * WAVES_PER_BLOCK + wave;
    if (b >= B) return;

    const int j0 = off1[b];
    const int j1 = off1[b + 1];

    const v4f* __restrict__ t4 = (const v4f*)table1;   // row r at t4[r*32]
    v4f acc = (v4f)0.0f;

    if (j0 < j1) {
        {
            int m = j1 - j0; if (m > CHUNK) m = CHUNK;
            if (lane < m)
                async_g2l_b32(lds_off(&sIdx[wave][0][lane]), &idx1[j0 + lane]);
        }
        int buf = 0;
        for (int jc = j0; jc < j1; jc += CHUNK, buf ^= 1) {
            int m = j1 - jc; if (m > CHUNK) m = CHUNK;
            const int jn = jc + CHUNK;
            if (jn < j1) {
                int mn = j1 - jn; if (mn > CHUNK) mn = CHUNK;
                if (lane < mn)
                    async_g2l_b32(lds_off(&sIdx[wave][buf ^ 1][lane]), &idx1[jn + lane]);
                asm volatile("s_wait_asynccnt 0x1" ::: "memory");
            } else {
                asm volatile("s_wait_asynccnt 0x0" ::: "memory");
            }

            if (lane < m) {
                const int pr = sIdx[wave][buf][lane];
                __builtin_prefetch(&t4[(size_t)pr * 32],      0, 0);  // row start
                __builtin_prefetch(&t4[(size_t)pr * 32 + 16], 0, 0);  // +256 B
            }

            int k = 0;
            for (; k + 8 <= m; k += 8) {
                v4f v[8];
                #pragma unroll
                for (int u = 0; u < 8; ++u) {
                    const int r = sIdx[wave][buf][k + u];
                    v[u] = t4[(size_t)r * 32 + lane];
                }
                #pragma unroll
                for (int u = 0; u < 8; ++u) acc += v[u];
            }
            for (; k < m; ++k) {
                const int r = sIdx[wave][buf][k];
                acc += t4[(size_t)r * 32 + lane];
            }
        }
    }
    const float inv = (j1 > j0) ? (1.0f / (float)(j1 - j0)) : 0.0f;  // empty -> patch 0
    acc *= inv;
    __builtin_nontemporal_store(acc, (v4f*)out1 + (size_t)b * 32 + lane);
}

// =============== fused launcher kernel ===============
__global__ __launch_bounds__(256) void fused_pool_kernel(
    const float* __restrict__ table0, const float* __restrict__ table1,
    const float* __restrict__ w0,
    const int* __restrict__ idx0, const int* __restrict__ idx1,
    const int* __restrict__ off0, const int* __restrict__ off1,
    float* __restrict__ out0, float* __restrict__ out1,
    int B, int blocks0)
{
    __shared__ int   sIdx[WAVES_PER_BLOCK][2][CHUNK];
    __shared__ float sW  [WAVES_PER_BLOCK][2][CHUNK];

    const int bid = (int)blockIdx.x;
    if (bid < blocks0) {
        pool0_body(bid, table0, w0, idx0, off0, out0, B, sIdx, sW);
    } else {
        pool1_body(bid - blocks0, table1, idx1, off1, out1, B, sIdx);
    }
}

extern "C" void kernel_launch(void* const* d_in, const int* in_sizes, int n_in,
                              void* d_out, int out_size, void* d_ws, size_t ws_size,
                              hipStream_t stream) {
    // setup_inputs order: table0, table1, weights0, idx0, idx1, off0, off1
    const float* table0 = (const float*)d_in[0];
    const float* table1 = (const float*)d_in[1];
    const float* w0     = (const float*)d_in[2];
    const int*   idx0   = (const int*)d_in[3];
    const int*   idx1   = (const int*)d_in[4];
    const int*   off0   = (const int*)d_in[5];
    const int*   off1   = (const int*)d_in[6];

    const int B = in_sizes[5] - 1;          // off0 has B+1 entries

    float* out  = (float*)d_out;
    float* out0 = out;                       // [B,64]
    float* out1 = out + (size_t)B * 64;      // [B,128]

    const int blocks0 = (B + WAVES_PER_BLOCK - 1) / WAVES_PER_BLOCK;
    const int total   = 2 * blocks0;         // second half handles feature 1

    fused_pool_kernel<<<total, 256, 0, stream>>>(table0, table1, w0,
                                                 idx0, idx1, off0, off1,
                                                 out0, out1, B, blocks0);
}